// RyokoRNN_420906795385
// MI455X (gfx1250) — compile-verified
//
#include <hip/hip_runtime.h>

#define L_   12
#define D_   1024
#define H_   4096
#define V_   50257
#define EPS_ 1e-5f
#define NBLK 256   // persistent grid; assumes co-residency (LDS ~28KB, 8 waves/block)

typedef float v2f __attribute__((ext_vector_type(2)));
typedef float v4f __attribute__((ext_vector_type(4)));
typedef float v8f __attribute__((ext_vector_type(8)));

struct RwkvP {
  const int*   ctx;
  const float* state;
  const float* emb;
  const float* ln0; const float* ln1; const float* ln2; const float* lnout;
  const float* tmk; const float* tmv; const float* tmr;
  const float* tfirst; const float* tdecay;
  const float* kw; const float* vw; const float* rw; const float* ow;
  const float* ftmk; const float* ftmr;
  const float* fkw; const float* fvw; const float* frw;
  float* out;                 // [V logits][5*L*D state]
  unsigned* bar;              // ws: {count, gen}
  float* xb0; float* xb1;     // ws: x ping-pong [D]
  float* wkv;                 // ws: [D]
  float* obuf;                // ws: ow@wkv [D]
  float* r2;                  // ws: [D]
  float* xf;                  // ws: final rms'ed x [D]
  float* k2;                  // ws: [H]
};

// ---------------- device helpers ----------------

__device__ __forceinline__ void grid_barrier(unsigned* cnt, unsigned* gen) {
  __threadfence();
  __syncthreads();
  if (threadIdx.x == 0) {
    unsigned g = __hip_atomic_load(gen, __ATOMIC_RELAXED, __HIP_MEMORY_SCOPE_AGENT);
    unsigned arrived = __hip_atomic_fetch_add(cnt, 1u, __ATOMIC_ACQ_REL, __HIP_MEMORY_SCOPE_AGENT);
    if (arrived == gridDim.x - 1) {
      __hip_atomic_store(cnt, 0u, __ATOMIC_RELAXED, __HIP_MEMORY_SCOPE_AGENT);
      __hip_atomic_fetch_add(gen, 1u, __ATOMIC_RELEASE, __HIP_MEMORY_SCOPE_AGENT);
    } else {
      while (__hip_atomic_load(gen, __ATOMIC_ACQUIRE, __HIP_MEMORY_SCOPE_AGENT) == g) {
        __builtin_amdgcn_s_sleep(2);
      }
    }
  }
  __syncthreads();
  __threadfence();
}

__device__ __forceinline__ float block_reduce_sum(float v, float* red) {
  #pragma unroll
  for (int off = 16; off > 0; off >>= 1) v += __shfl_xor(v, off, 32);
  if ((threadIdx.x & 31) == 0) red[threadIdx.x >> 5] = v;
  __syncthreads();
  float s = 0.f;
  #pragma unroll
  for (int j = 0; j < 8; ++j) s += red[j];
  __syncthreads();
  return s;
}

// Async memory->LDS stage: 256 threads x b128 per pass (4KB/pass), ASYNCcnt
// tracked (GLOBAL_LOAD_ASYNC_TO_LDS_B128, GVS mode: lds-addr VGPR, u32 voffset,
// SGPR base). nfloats % 1024 == 0. Includes wait + block barrier.
__device__ __forceinline__ void async_stage_to_lds(float* lds_dst,
                                                   const float* __restrict__ gsrc,
                                                   int nfloats) {
  const int tid = threadIdx.x;
  for (int base = 0; base < nfloats; base += 1024) {
    unsigned ldsa = (unsigned)(size_t)(lds_dst + base + tid * 4);   // low32 = LDS offset
    unsigned voff = (unsigned)((base + tid * 4) * sizeof(float));
    asm volatile("global_load_async_to_lds_b128 %0, %1, %2"
                 :: "v"(ldsa), "v"(voff), "s"(gsrc)
                 : "memory");
  }
  asm volatile("s_wait_asynccnt 0x0" ::: "memory");
  __syncthreads();
}

// One wave computes dot(row[0..K), bv[0..K)); K % 1024 == 0.
// Streaming non-temporal 16B loads, 8 in flight, coalesced 512B/instr per wave.
__device__ __forceinline__ float wave_dot(const float* __restrict__ row,
                                          const float* __restrict__ bv, int K) {
  const int lane = threadIdx.x & 31;
  float acc = 0.f;
  for (int base = lane * 4; base < K; base += 1024) {
    if (base + 1024 < K) __builtin_prefetch(row + base + 1024, 0, 1); // global_prefetch_b8
    v4f a[8], b[8];
    #pragma unroll
    for (int u = 0; u < 8; ++u)
      a[u] = __builtin_nontemporal_load((const v4f*)(row + base + u * 128));
    #pragma unroll
    for (int u = 0; u < 8; ++u)
      b[u] = *(const v4f*)(bv + base + u * 128);
    #pragma unroll
    for (int u = 0; u < 8; ++u) {
      v4f m = a[u] * b[u];
      acc += m.x + m.y + m.z + m.w;
    }
  }
  #pragma unroll
  for (int off = 16; off > 0; off >>= 1) acc += __shfl_xor(acc, off, 32);
  return acc; // all lanes hold the result
}

__device__ __forceinline__ float sigmoidf_(float x) { return 1.f / (1.f + expf(-x)); }

// ---------------- persistent layer kernel ----------------

__global__ void __launch_bounds__(256)
rwkv_layers_kernel(RwkvP p) {
  __shared__ __align__(16) float ls0[D_];
  __shared__ __align__(16) float ls1[D_];
  __shared__ __align__(16) float ls2[D_];
  __shared__ __align__(16) float ls_h[H_];
  __shared__ float red_s[8];

  const int tid   = threadIdx.x;
  const int lane  = tid & 31;
  const int wv    = tid >> 5;
  const int gwave = blockIdx.x * 8 + wv;
  const int NW    = gridDim.x * 8;
  float* so = p.out + V_;   // new-state output base

  // ---- pre-stage: x0 = rms(emb[ctx[-1]], ln0); block 0 persists to xb0 ----
  {
    const int tok = p.ctx[0];
    const float* er = p.emb + (size_t)tok * D_;
    float xr_[4]; float ss = 0.f;
    #pragma unroll
    for (int u = 0; u < 4; ++u) { int i = tid + u * 256; float v = er[i]; xr_[u] = v; ss += v * v; }
    ss = block_reduce_sum(ss, red_s);
    float rq = rsqrtf(ss * (1.f / D_) + EPS_);
    if (blockIdx.x == 0) {
      #pragma unroll
      for (int u = 0; u < 4; ++u) { int i = tid + u * 256; p.xb0[i] = p.ln0[i] * xr_[u] * rq; }
    }
  }
  grid_barrier(&p.bar[0], &p.bar[1]);

  float* xin  = p.xb0;
  float* xout = p.xb1;
  float pend  = 1.f;   // deferred residual-rescale factor from previous layer

  for (int l = 0; l < L_; ++l) {
    const float* l1  = p.ln1    + l * D_;
    const float* l2  = p.ln2    + l * D_;
    const float* tmk = p.tmk    + l * D_;
    const float* tmv = p.tmv    + l * D_;
    const float* tmr = p.tmr    + l * D_;
    const float* tf  = p.tfirst + l * D_;
    const float* td  = p.tdecay + l * D_;
    const float* ftk = p.ftmk   + l * D_;
    const float* ftr = p.ftmr   + l * D_;
    const float* s_ff = p.state + (size_t)(l * 5 + 0) * D_;
    const float* s_sa = p.state + (size_t)(l * 5 + 1) * D_;
    const float* s_aa = p.state + (size_t)(l * 5 + 2) * D_;
    const float* s_bb = p.state + (size_t)(l * 5 + 3) * D_;
    const float* s_pp = p.state + (size_t)(l * 5 + 4) * D_;

    // ===== Stage A: xx=rms(x,l1); xk/xv/xr (block-redundant, into LDS);
    //               then per-channel waves: r/k/v dots + fused WKV state update
    {
      float xs[4]; float ss = 0.f;
      #pragma unroll
      for (int u = 0; u < 4; ++u) { int i = tid + u * 256; float v = xin[i] * pend; xs[u] = v; ss += v * v; }
      ss = block_reduce_sum(ss, red_s);
      float rq = rsqrtf(ss * (1.f / D_) + EPS_);
      #pragma unroll
      for (int u = 0; u < 4; ++u) {
        int i = tid + u * 256;
        float xx = l1[i] * xs[u] * rq;
        float sa = s_sa[i];
        ls0[i] = xx * tmk[i] + sa * (1.f - tmk[i]);   // xk
        ls1[i] = xx * tmv[i] + sa * (1.f - tmv[i]);   // xv
        ls2[i] = xx * tmr[i] + sa * (1.f - tmr[i]);   // xr
        if (blockIdx.x == 0) so[(size_t)(l * 5 + 1) * D_ + i] = xx;  // new sa_x
      }
      __syncthreads();

      for (int i = gwave; i < D_; i += NW) {
        float dk = wave_dot(p.kw + ((size_t)l * D_ + i) * D_, ls0, D_);
        float dv = wave_dot(p.vw + ((size_t)l * D_ + i) * D_, ls1, D_);
        float dr = wave_dot(p.rw + ((size_t)l * D_ + i) * D_, ls2, D_);
        if (lane == 0) {
          float r  = sigmoidf_(dr);
          float aa = s_aa[i], bb = s_bb[i], pp = s_pp[i];
          float ww = tf[i] + dk;
          float pm = fmaxf(pp, ww);
          float e1 = expf(pp - pm), e2 = expf(ww - pm);
          float a  = e1 * aa + e2 * dv;
          float b  = e1 * bb + e2;
          p.wkv[i] = r * (a / b);
          float ww2 = pp + td[i];
          float p2  = fmaxf(ww2, dk);
          float e1b = expf(ww2 - p2), e2b = expf(dk - p2);
          so[(size_t)(l * 5 + 2) * D_ + i] = e1b * aa + e2b * dv;  // naa
          so[(size_t)(l * 5 + 3) * D_ + i] = e1b * bb + e2b;       // nbb
          so[(size_t)(l * 5 + 4) * D_ + i] = p2;                   // pp
        }
      }
    }
    grid_barrier(&p.bar[0], &p.bar[1]);

    // ===== Stage B: obuf = ow @ wkv  (wkv staged to LDS via async DMA)
    {
      async_stage_to_lds(ls0, p.wkv, D_);
      for (int r = gwave; r < D_; r += NW) {
        float d = wave_dot(p.ow + ((size_t)l * D_ + r) * D_, ls0, D_);
        if (lane == 0) p.obuf[r] = d;
      }
    }
    grid_barrier(&p.bar[0], &p.bar[1]);

    // ===== Stage C: xc = x + obuf; yy=rms(xc,l2); xk2/xr2 (LDS);
    //               k2 = relu(fkw@xk2)^2 ; r2 = sigmoid(frw@xr2)
    {
      float xs[4]; float ss = 0.f;
      #pragma unroll
      for (int u = 0; u < 4; ++u) {
        int i = tid + u * 256;
        float v = xin[i] * pend + p.obuf[i];
        xs[u] = v; ss += v * v;
      }
      ss = block_reduce_sum(ss, red_s);
      float rq = rsqrtf(ss * (1.f / D_) + EPS_);
      #pragma unroll
      for (int u = 0; u < 4; ++u) {
        int i = tid + u * 256;
        float yy = l2[i] * xs[u] * rq;
        float fx = s_ff[i];
        ls0[i] = yy * ftk[i] + fx * (1.f - ftk[i]);   // xk2
        ls1[i] = yy * ftr[i] + fx * (1.f - ftr[i]);   // xr2
        if (blockIdx.x == 0) {
          so[(size_t)(l * 5 + 0) * D_ + i] = yy;      // new ff_x
          xout[i] = xs[u];                            // persist xc
        }
      }
      __syncthreads();

      for (int t = gwave; t < H_ + D_; t += NW) {
        if (t < H_) {
          float d = wave_dot(p.fkw + ((size_t)l * H_ + t) * D_, ls0, D_);
          if (lane == 0) { float rl = fmaxf(d, 0.f); p.k2[t] = rl * rl; }
        } else {
          int r = t - H_;
          float d = wave_dot(p.frw + ((size_t)l * D_ + r) * D_, ls1, D_);
          if (lane == 0) p.r2[r] = sigmoidf_(d);
        }
      }
    }
    grid_barrier(&p.bar[0], &p.bar[1]);

    // ===== Stage D: x = xc + r2 * (fvw @ k2)   (k2 staged via async DMA)
    {
      async_stage_to_lds(ls_h, p.k2, H_);
      for (int r = gwave; r < D_; r += NW) {
        float d = wave_dot(p.fvw + ((size_t)l * D_ + r) * H_, ls_h, H_);
        if (lane == 0) xout[r] = xout[r] + p.r2[r] * d;
      }
    }
    grid_barrier(&p.bar[0], &p.bar[1]);

    pend = (((l + 1) % 6) == 0) ? 0.5f : 1.f;
    float* t = xin; xin = xout; xout = t;
  }

  // ---- final: xf = rms(x * pend, lnout), block 0 only ----
  if (blockIdx.x == 0) {
    float xs[4]; float ss = 0.f;
    #pragma unroll
    for (int u = 0; u < 4; ++u) { int i = tid + u * 256; float v = xin[i] * pend; xs[u] = v; ss += v * v; }
    ss = block_reduce_sum(ss, red_s);
    float rq = rsqrtf(ss * (1.f / D_) + EPS_);
    #pragma unroll
    for (int u = 0; u < 4; ++u) { int i = tid + u * 256; p.xf[i] = p.lnout[i] * xs[u] * rq; }
  }
}

// ---------------- head GEMV via V_WMMA_F32_16X16X4_F32 ----------------
// Each wave produces 16 logits rows. A-fragment (16x4 fp32): lanes 0-15 hold
// K=0,1 in v[0..1]; lanes 16-31 hold K=2,3 (ISA 7.12.2). B column 0 = x chunk:
// VGPR0 lanes0-15 = B[0][n], lanes16-31 = B[1][n]; VGPR1 = rows 2,3.
// Result column 0: lane0 v[0..7] = rows 0..7, lane16 v[0..7] = rows 8..15.
__global__ void __launch_bounds__(256)
rwkv_head_kernel(const float* __restrict__ headw, const float* __restrict__ xf,
                 float* __restrict__ logits) {
  __shared__ __align__(16) float bx[D_];
  async_stage_to_lds(bx, xf, D_);   // async DMA x into LDS (includes barrier)

  const int wv   = threadIdx.x >> 5;
  const int lane = threadIdx.x & 31;
  const int grp  = blockIdx.x * 8 + wv;
  const int G    = (V_ + 15) >> 4;      // 3142 row-groups
  if (grp >= G) return;                 // uniform per wave -> EXEC all-ones below

  const int half = lane >> 4;           // 0: K=0,1 ; 1: K=2,3
  const int m    = lane & 15;
  int row  = grp * 16 + m;
  int rowc = row < V_ ? row : V_ - 1;   // clamp loads for the 1-row tail group
  const float* rp = headw + (size_t)rowc * D_ + half * 2;

  v8f c = {};
  #pragma unroll 8
  for (int k = 0; k < D_; k += 4) {
    if ((k & 255) == 0) __builtin_prefetch(rp + k + 256, 0, 1);
    v2f a = __builtin_nontemporal_load((const v2f*)(rp + k));
    v2f b;
    b.x = bx[k + half];
    b.y = bx[k + 2 + half];
    c = __builtin_amdgcn_wmma_f32_16x16x4_f32(false, a, false, b, (short)0, c,
                                              false, false);
  }

  if (m == 0) {                          // lanes 0 and 16 hold column 0
    int rbase = grp * 16 + half * 8;
    #pragma unroll
    for (int j = 0; j < 8; ++j)
      if (rbase + j < V_) logits[rbase + j] = c[j];
  }
}

// ---------------- launch ----------------

extern "C" void kernel_launch(void* const* d_in, const int* in_sizes, int n_in,
                              void* d_out, int out_size, void* d_ws, size_t ws_size,
                              hipStream_t stream) {
  (void)in_sizes; (void)n_in; (void)out_size; (void)ws_size;

  RwkvP p;
  p.ctx    = (const int*)  d_in[0];
  p.state  = (const float*)d_in[1];
  p.emb    = (const float*)d_in[2];
  p.ln0    = (const float*)d_in[3];
  p.ln1    = (const float*)d_in[4];
  p.ln2    = (const float*)d_in[5];
  p.lnout  = (const float*)d_in[6];
  p.tmk    = (const float*)d_in[7];
  p.tmv    = (const float*)d_in[8];
  p.tmr    = (const float*)d_in[9];
  p.tfirst = (const float*)d_in[10];
  p.tdecay = (const float*)d_in[11];
  p.kw     = (const float*)d_in[12];
  p.vw     = (const float*)d_in[13];
  p.rw     = (const float*)d_in[14];
  p.ow     = (const float*)d_in[15];
  p.ftmk   = (const float*)d_in[16];
  p.ftmr   = (const float*)d_in[17];
  p.fkw    = (const float*)d_in[18];
  p.fvw    = (const float*)d_in[19];
  p.frw    = (const float*)d_in[20];
  const float* headw = (const float*)d_in[21];
  p.out = (float*)d_out;

  unsigned char* w = (unsigned char*)d_ws;
  p.bar  = (unsigned*)w;                 // 2 u32, padded to 256B
  float* f = (float*)(w + 256);
  p.xb0  = f;            f += D_;
  p.xb1  = f;            f += D_;
  p.wkv  = f;            f += D_;
  p.obuf = f;            f += D_;
  p.r2   = f;            f += D_;
  p.xf   = f;            f += D_;
  p.k2   = f;            f += H_;

  // reset grid-barrier state (graph-capture safe)
  hipMemsetAsync(d_ws, 0, 256, stream);

  rwkv_layers_kernel<<<NBLK, 256, 0, stream>>>(p);

  const int G = (V_ + 15) >> 4;          // 3142 groups of 16 rows
  rwkv_head_kernel<<<(G + 7) / 8, 256, 0, stream>>>(headw, p.xf, (float*)d_out);
}